// ProbAttention_53815940219424
// MI455X (gfx1250) — compile-verified
//
#include <hip/hip_runtime.h>
#include <math.h>

typedef float v2f __attribute__((ext_vector_type(2)));
typedef float v8f __attribute__((ext_vector_type(8)));

#define B_ 2
#define H_ 8
#define L_ 4096
#define D_ 64
#define U_ 45      // sample_k and top-u (both = 45 for these shapes)
#define UPAD_ 64   // 4 tiles of 16 rows
#define SCALE_ 0.125f   // 1/sqrt(64)

// ---------------------------------------------------------------------------
// Cross-lane max over each 16-lane row group via DPP16 ROW_ROR rotates
// (pure VALU, avoids ds_bpermute round-trips through LDS hardware).
// ---------------------------------------------------------------------------
__device__ __forceinline__ float rowmax16(float x) {
    float t;
    t = __int_as_float(__builtin_amdgcn_update_dpp(
            0, __float_as_int(x), 0x121, 0xF, 0xF, true)); // ROW_ROR:1
    x = fmaxf(x, t);
    t = __int_as_float(__builtin_amdgcn_update_dpp(
            0, __float_as_int(x), 0x122, 0xF, 0xF, true)); // ROW_ROR:2
    x = fmaxf(x, t);
    t = __int_as_float(__builtin_amdgcn_update_dpp(
            0, __float_as_int(x), 0x124, 0xF, 0xF, true)); // ROW_ROR:4
    x = fmaxf(x, t);
    t = __int_as_float(__builtin_amdgcn_update_dpp(
            0, __float_as_int(x), 0x128, 0xF, 0xF, true)); // ROW_ROR:8
    x = fmaxf(x, t);
    return x;
}

// ---------------------------------------------------------------------------
// Stage 1: M[b,h,q] = max_s(q . K[idx[q,s]]) - sum_s(...)/L
// ---------------------------------------------------------------------------
__global__ void probe_kernel(const float* __restrict__ Q,
                             const float* __restrict__ K,
                             const int*   __restrict__ idxs,
                             float* __restrict__ Mout, int sk) {
    int t = blockIdx.x * blockDim.x + threadIdx.x;   // t = bh*L + q
    if (t >= B_ * H_ * L_) return;
    int q  = t & (L_ - 1);
    int bh = t >> 12;            // L_ = 4096 = 2^12
    const float* qp = Q + (size_t)t * D_;
    float qreg[D_];
    #pragma unroll
    for (int i = 0; i < D_; ++i) qreg[i] = qp[i];
    float mx = -3.0e38f, sm = 0.f;
    for (int s = 0; s < sk; ++s) {
        int ki = idxs[q * sk + s];
        const float* kp = K + ((size_t)bh * L_ + ki) * D_;
        float acc = 0.f;
        #pragma unroll
        for (int i = 0; i < D_; ++i) acc = fmaf(qreg[i], kp[i], acc);
        mx = fmaxf(mx, acc);
        sm += acc;
    }
    Mout[t] = mx - sm * (1.0f / (float)L_);
}

// ---------------------------------------------------------------------------
// Stage 2: per (b,h) iterative top-45 selection (tie-break: lowest index)
// ---------------------------------------------------------------------------
__global__ void topk_kernel(const float* __restrict__ M, int* __restrict__ Mtop) {
    int bh  = blockIdx.x;
    int tid = threadIdx.x;                 // 256 threads
    __shared__ float sM[L_];
    __shared__ float bestV[256];
    __shared__ int   bestI[256];
    for (int i = tid; i < L_; i += 256) sM[i] = M[bh * L_ + i];
    __syncthreads();
    for (int it = 0; it < UPAD_; ++it) {
        float bv = -3.0e38f; int bi = 0;
        if (it < U_) {
            for (int i = tid; i < L_; i += 256) {
                float v = sM[i];
                if (v > bv || (v == bv && i < bi)) { bv = v; bi = i; }
            }
        }
        bestV[tid] = bv; bestI[tid] = bi;
        __syncthreads();
        for (int s = 128; s > 0; s >>= 1) {
            if (tid < s) {
                float v2 = bestV[tid + s]; int i2 = bestI[tid + s];
                if (v2 > bestV[tid] || (v2 == bestV[tid] && i2 < bestI[tid])) {
                    bestV[tid] = v2; bestI[tid] = i2;
                }
            }
            __syncthreads();
        }
        if (tid == 0) {
            int sel = (it < U_) ? bestI[0] : 0;
            Mtop[bh * UPAD_ + it] = sel;
            if (it < U_) sM[sel] = -3.0e38f;
        }
        __syncthreads();
    }
}

// ---------------------------------------------------------------------------
// Stage 3: context baseline = cumsum(V, axis=L). One block per (b,h).
// ---------------------------------------------------------------------------
__global__ void cumsum_kernel(const float* __restrict__ V, float* __restrict__ out) {
    int bh  = blockIdx.x;
    int d   = threadIdx.x & 63;
    int seg = threadIdx.x >> 6;           // 0..3, 1024 rows each
    __shared__ float partial[4][64];
    const float* vb = V   + (size_t)bh * L_ * D_;
    float*       ob = out + (size_t)bh * L_ * D_;
    int l0 = seg * 1024;
    float acc = 0.f;
    for (int l = l0; l < l0 + 1024; ++l) acc += vb[l * D_ + d];
    partial[seg][d] = acc;
    __syncthreads();
    float run = 0.f;
    for (int s2 = 0; s2 < seg; ++s2) run += partial[s2][d];
    for (int l = l0; l < l0 + 1024; ++l) {
        run += vb[l * D_ + d];
        ob[l * D_ + d] = run;
    }
}

// ---------------------------------------------------------------------------
// Stage 4: masked attention for the top-45 queries, fp32 WMMA flash-style.
// One block (128 thr = 4 wave32) per (b,h); wave w owns M-tile w (16 rows).
// Row sums of P computed by WMMA itself (ones-column tile); row max via DPP.
// ---------------------------------------------------------------------------
__global__ void __launch_bounds__(128)
attn_kernel(const float* __restrict__ Q, const float* __restrict__ K,
            const float* __restrict__ V, const int* __restrict__ Mtop,
            float* __restrict__ out) {
    int bh   = blockIdx.x;
    int tid  = threadIdx.x;
    int wave = tid >> 5;
    int lane = tid & 31;
    int half = lane >> 4;       // 0: lanes 0-15, 1: lanes 16-31
    int l16  = lane & 15;

    __shared__ float sQ[UPAD_][D_ + 1];   // +1 pad: conflict-free column reads
    __shared__ float sK[16][D_ + 1];
    __shared__ float sV[16][D_ + 1];
    __shared__ float sP[4][16][17];       // per-wave P tile (C-layout -> A-layout)
    __shared__ int   sTop[UPAD_];

    for (int i = tid; i < UPAD_; i += 128) sTop[i] = Mtop[bh * UPAD_ + i];
    __syncthreads();
    for (int idx = tid; idx < UPAD_ * D_; idx += 128) {
        int row = idx >> 6, col = idx & 63;
        float v = 0.f;
        if (row < U_) v = Q[((size_t)bh * L_ + sTop[row]) * D_ + col];
        sQ[row][col] = v;
    }
    __syncthreads();

    // Preload A-fragments of Q (ISA 16x4 f32 layout: lanes0-15 K={0,1}, lanes16-31 K={2,3})
    int arow = wave * 16 + l16;
    v2f aq[16];
    #pragma unroll
    for (int kk = 0; kk < 16; ++kk) {
        aq[kk].x = sQ[arow][4 * kk + 2 * half + 0];
        aq[kk].y = sQ[arow][4 * kk + 2 * half + 1];
    }
    // per-lane row state (C-layout: VGPR r holds row r + 8*half for this lane)
    int   topv[8];
    float m_run[8];
    #pragma unroll
    for (int r = 0; r < 8; ++r) {
        topv[r]  = sTop[wave * 16 + r + 8 * half];
        m_run[r] = -3.0e38f;
    }
    v8f Oacc[4], Osum;
    #pragma unroll
    for (int r = 0; r < 8; ++r) {
        Osum[r] = 0.f;
        #pragma unroll
        for (int dt = 0; dt < 4; ++dt) Oacc[dt][r] = 0.f;
    }
    // constant B-fragment for the "ones column" tile: B[c][n] = (n==0)
    v2f bones;
    bones.x = (l16 == 0) ? 1.0f : 0.0f;
    bones.y = bones.x;

    const float* Kb = K + (size_t)bh * L_ * D_;
    const float* Vb = V + (size_t)bh * L_ * D_;

    for (int kb = 0; kb < L_ / 16; ++kb) {
        __syncthreads();   // previous tile fully consumed
        int kbase = kb * 16;
        for (int idx = tid; idx < 16 * D_; idx += 128) {
            int row = idx >> 6, col = idx & 63;
            sK[row][col] = Kb[(size_t)(kbase + row) * D_ + col];
            sV[row][col] = Vb[(size_t)(kbase + row) * D_ + col];
        }
        __syncthreads();

        // ---- S = Q * K^T : batch all 16 B-fragment loads, then dense WMMA chain
        v2f bk[16];
        #pragma unroll
        for (int kk = 0; kk < 16; ++kk) {
            bk[kk].x = sK[l16][4 * kk + 2 * half + 0];
            bk[kk].y = sK[l16][4 * kk + 2 * half + 1];
        }
        v8f s = {0.f, 0.f, 0.f, 0.f, 0.f, 0.f, 0.f, 0.f};
        #pragma unroll
        for (int kk = 0; kk < 16; ++kk) {
            s = __builtin_amdgcn_wmma_f32_16x16x4_f32(
                    false, aq[kk], false, bk[kk], (short)0, s, false, false);
        }

        // ---- scale + mask + online softmax (max via DPP rotates, sum via WMMA)
        int ncol = kbase + l16;
        #pragma unroll
        for (int r = 0; r < 8; ++r) {
            float x = s[r] * SCALE_;
            x = (ncol > topv[r]) ? -1.0e30f : x;
            float mx   = rowmax16(x);
            float mnew = fmaxf(m_run[r], mx);
            float corr = __expf(m_run[r] - mnew);
            m_run[r]   = mnew;
            float p    = __expf(x - mnew);
            sP[wave][r + 8 * half][l16] = p;   // P tile to LDS (C-layout)
            Osum[r] *= corr;
            #pragma unroll
            for (int dt = 0; dt < 4; ++dt) Oacc[dt][r] *= corr;
        }
        asm volatile("s_wait_dscnt 0" ::: "memory");  // same-wave DS RAW

        // ---- O += P * V (A = P in 4 K-chunks); 5th tile accumulates row sums
        v2f ap[4];
        #pragma unroll
        for (int cc = 0; cc < 4; ++cc) {
            ap[cc].x = sP[wave][l16][4 * cc + 2 * half + 0];
            ap[cc].y = sP[wave][l16][4 * cc + 2 * half + 1];
        }
        #pragma unroll
        for (int cc = 0; cc < 4; ++cc) {
            v2f bv[4];
            #pragma unroll
            for (int dt = 0; dt < 4; ++dt) {
                bv[dt].x = sV[4 * cc + 2 * half + 0][dt * 16 + l16];
                bv[dt].y = sV[4 * cc + 2 * half + 1][dt * 16 + l16];
            }
            #pragma unroll
            for (int dt = 0; dt < 4; ++dt) {
                Oacc[dt] = __builtin_amdgcn_wmma_f32_16x16x4_f32(
                               false, ap[cc], false, bv[dt], (short)0, Oacc[dt], false, false);
            }
            Osum = __builtin_amdgcn_wmma_f32_16x16x4_f32(
                       false, ap[cc], false, bones, (short)0, Osum, false, false);
        }
    }

    // epilogue: row sum lives in lanes with l16==0 of each half-group; broadcast,
    // normalize, scatter rows into cumsum baseline
    #pragma unroll
    for (int r = 0; r < 8; ++r) {
        float lsum = __shfl(Osum[r], half * 16, 32);
        float inv  = 1.0f / lsum;
        int grow = wave * 16 + r + 8 * half;
        if (grow < U_) {
            #pragma unroll
            for (int dt = 0; dt < 4; ++dt) {
                out[((size_t)bh * L_ + topv[r]) * D_ + dt * 16 + l16] =
                    Oacc[dt][r] * inv;
            }
        }
    }
}

// ---------------------------------------------------------------------------
extern "C" void kernel_launch(void* const* d_in, const int* in_sizes, int n_in,
                              void* d_out, int out_size, void* d_ws, size_t ws_size,
                              hipStream_t stream) {
    const float* Q    = (const float*)d_in[0];
    const float* K    = (const float*)d_in[1];
    const float* V    = (const float*)d_in[2];
    const int*   idxs = (const int*)d_in[3];
    float* out = (float*)d_out;

    float* Mscore = (float*)d_ws;                                     // B*H*L floats
    int*   Mtop   = (int*)((char*)d_ws + (size_t)B_ * H_ * L_ * 4);   // B*H*64 ints
    int sk = in_sizes[3] / L_;   // 45 for these shapes

    probe_kernel <<<(B_ * H_ * L_) / 256, 256, 0, stream>>>(Q, K, idxs, Mscore, sk);
    topk_kernel  <<<B_ * H_, 256, 0, stream>>>(Mscore, Mtop);
    cumsum_kernel<<<B_ * H_, 256, 0, stream>>>(V, out);
    attn_kernel  <<<B_ * H_, 128, 0, stream>>>(Q, K, V, Mtop, out);
}